// MaskEmbeddingLoss_51247549776583
// MI455X (gfx1250) — compile-verified
//
#include <hip/hip_runtime.h>
#include <math.h>

#define NPTS   500000
#define NCLUST 16
#define PADN   (1u << 19)          // 524288 >= NPTS, power of two for bitonic
#define NBLK   (PADN / 256)        // 2048 blocks of 256 sorted elements
#define TILE   1024u               // LDS-fused bitonic tile (4 KB per block)

typedef __attribute__((ext_vector_type(2))) float v2f;
typedef __attribute__((ext_vector_type(8))) float v8f;

// -------- workspace layout (element offsets) --------
// float region:
//   [0,80)        accum[c][5]   : cnt, ex, ey, ez, margin_sum
//   [128,192)     centroids padded [c][4] (4th = 0 for WMMA K=4)
//   [192,208)     c2[c]
//   [208,224)     sigma[c]
//   [224,240)     inv2s2[c] = 1/(2*sigma^2)
//   [240,256)     countsF[c]
//   [256]         reg_loss
//   [1024, 1024+NPTS)  probs[n]
// u32 region:
//   [524288, 524288+16*PADN)     sort keys (cluster-major)
//   [8912896, +NBLK*16)          per-block gt counts
//   [8945664, +NBLK*16)          exclusive-scanned gt bases
// float region:
//   [8978432, +NBLK*16)          per-block lovasz contribution partials
#define OFF_ACCUM   0
#define OFF_CENT    128
#define OFF_C2      192
#define OFF_SIGMA   208
#define OFF_INV2S2  224
#define OFF_CNTF    240
#define OFF_REG     256
#define OFF_PROBS   1024
#define OFF_KEYS    524288u
#define OFF_GTP     8912896u
#define OFF_BASE    8945664u
#define OFF_LPART   8978432u

// ---------------------------------------------------------------------------
// K1: per-cluster segmented reductions (deterministic: one block per cluster,
// fixed-order LDS tree).
// ---------------------------------------------------------------------------
__global__ void k_cluster_stats(const float* __restrict__ emb,
                                const float* __restrict__ marg,
                                const int*   __restrict__ lab,
                                float*       __restrict__ wsf)
{
    const int c   = blockIdx.x;
    const int tid = threadIdx.x;
    float v[5] = {0.f, 0.f, 0.f, 0.f, 0.f};
    for (int n = tid; n < NPTS; n += blockDim.x) {
        // speculative prefetch a few strides ahead (global_prefetch_b8)
        __builtin_prefetch(&emb[3 * (n + 8 * 256)], 0, 1);
        if (lab[n] == c) {
            v[0] += 1.0f;
            v[1] += emb[3 * n + 0];
            v[2] += emb[3 * n + 1];
            v[3] += emb[3 * n + 2];
            v[4] += marg[n];
        }
    }
    __shared__ float rf[256];
    for (int k = 0; k < 5; ++k) {
        rf[tid] = v[k];
        __syncthreads();
        for (int off = 128; off > 0; off >>= 1) {
            if (tid < off) rf[tid] += rf[tid + off];
            __syncthreads();
        }
        if (tid == 0) wsf[OFF_ACCUM + c * 5 + k] = rf[0];
        __syncthreads();
    }
}

// ---------------------------------------------------------------------------
// K2: finalize centroids, c2, sigma, 1/(2 sigma^2), counts, reg_loss.
// ---------------------------------------------------------------------------
__global__ void k_finalize_stats(float* __restrict__ wsf)
{
    const int tid = threadIdx.x;
    __shared__ float norms[NCLUST];
    if (tid < NCLUST) {
        const float* acc = wsf + OFF_ACCUM;
        float cnt = acc[tid * 5 + 0];
        float inv = 1.0f / cnt;
        float cx = acc[tid * 5 + 1] * inv;
        float cy = acc[tid * 5 + 2] * inv;
        float cz = acc[tid * 5 + 3] * inv;
        float sg = acc[tid * 5 + 4] * inv;
        wsf[OFF_CENT + tid * 4 + 0] = cx;
        wsf[OFF_CENT + tid * 4 + 1] = cy;
        wsf[OFF_CENT + tid * 4 + 2] = cz;
        wsf[OFF_CENT + tid * 4 + 3] = 0.0f;          // K=4 pad for WMMA
        wsf[OFF_C2 + tid]     = cx * cx + cy * cy + cz * cz;
        wsf[OFF_SIGMA + tid]  = sg;
        wsf[OFF_INV2S2 + tid] = 1.0f / (2.0f * sg * sg);
        wsf[OFF_CNTF + tid]   = cnt;
        float nx = cx + 1e-8f, ny = cy + 1e-8f, nz = cz + 1e-8f;
        norms[tid] = sqrtf(nx * nx + ny * ny + nz * nz);
    }
    __syncthreads();
    if (tid == 0) {
        float s = 0.f;
        for (int c = 0; c < NCLUST; ++c) s += norms[c];
        wsf[OFF_REG] = s / (float)NCLUST;
    }
}

// ---------------------------------------------------------------------------
// K3: WMMA distance tile kernel. One V_WMMA_F32_16X16X4_F32 per 16-point tile
// computes all 16 cluster dot products. Each lane then finishes p, the packed
// lovasz sort key (bits(err)<<1 | gt), and probs[n] for its 8 clusters.
// ---------------------------------------------------------------------------
__global__ void k_wmma_probs(const float* __restrict__ emb,
                             const int*   __restrict__ lab,
                             float*       __restrict__ wsf,
                             unsigned*    __restrict__ wsu)
{
    const int tid   = threadIdx.x;
    const int lane  = tid & 31;
    const int wv    = tid >> 5;
    const int tile  = blockIdx.x * 8 + wv;
    const int n     = tile * 16 + (lane & 15);
    const int khalf = lane >> 4;                  // 0: K={0,1}; 1: K={2,3}

    const float* cent = wsf + OFF_CENT;

    // A: 16x4 centroids (M = lane&15, K = 2*khalf + {0,1})
    v2f a;
    {
        int m = lane & 15;
        a.x = cent[m * 4 + 2 * khalf + 0];
        a.y = cent[m * 4 + 2 * khalf + 1];
    }
    // B: 4x16 tile of emb^T (N = lane&15, K = 2*khalf + {0,1}), K=3 padded 0
    v2f b;
    float e2  = 0.0f;
    int   myl = -1;
    if (n < NPTS) {
        float e0 = emb[3 * n + 0];
        float e1 = emb[3 * n + 1];
        float e2v = emb[3 * n + 2];
        b.x = khalf ? e2v : e0;
        b.y = khalf ? 0.0f : e1;
        e2  = e0 * e0 + e1 * e1 + e2v * e2v;
        myl = lab[n];
    } else {
        b.x = 0.0f;
        b.y = 0.0f;
    }

    v8f acc = {};
    acc = __builtin_amdgcn_wmma_f32_16x16x4_f32(
        /*neg_a=*/false, a, /*neg_b=*/false, b,
        /*c_mod=*/(short)0, acc, /*reuse_a=*/false, /*reuse_b=*/false);

    unsigned*    keys   = wsu + OFF_KEYS;
    const float* c2     = wsf + OFF_C2;
    const float* inv2s2 = wsf + OFF_INV2S2;
    float*       probs  = wsf + OFF_PROBS;

#pragma unroll
    for (int r = 0; r < 8; ++r) {
        int m = r + 8 * khalf;                    // D layout: M = r + 8*(lane>=16)
        unsigned key = 0u;
        if (n < NPTS) {
            float dot  = acc[r];
            float dist = c2[m] + e2 - 2.0f * dot;
            float p    = expf(-dist * inv2s2[m]);
            p          = fminf(fmaxf(p, 0.0f), 1.0f);
            int   gt   = (myl == m) ? 1 : 0;
            float err  = gt ? (2.0f - 2.0f * p) : (2.0f * p);   // 1 - (2p-1)*sign
            key = (__float_as_uint(err) << 1) | (unsigned)gt;   // err<=2 -> fits
            if (gt) probs[n] = p;
        }
        keys[(unsigned)m * PADN + (unsigned)n] = key;           // pad -> key 0
    }
}

// ---------------------------------------------------------------------------
// K4a: fused LDS bitonic sort of each 1024-element tile (stages k=2..TILE).
// Direction bits use the *global* element index, so this performs exactly the
// same comparator network as the global passes would.
// ---------------------------------------------------------------------------
__global__ void k_bitonic_local_sort(unsigned* __restrict__ keys)
{
    const unsigned c     = blockIdx.y;
    const unsigned gbase = blockIdx.x * TILE;
    unsigned* base = keys + (size_t)c * PADN;
    __shared__ unsigned s[TILE];
    for (unsigned e = threadIdx.x; e < TILE; e += blockDim.x)
        s[e] = base[gbase + e];
    __syncthreads();
    for (unsigned k = 2; k <= TILE; k <<= 1) {
        for (unsigned j = k >> 1; j > 0; j >>= 1) {
            for (unsigned t = threadIdx.x; t < TILE / 2; t += blockDim.x) {
                unsigned i = ((t & ~(j - 1)) << 1) | (t & (j - 1));
                unsigned p = i | j;
                unsigned va = s[i], vb = s[p];
                bool desc = (((gbase + i) & k) == 0);
                if (desc ? (va < vb) : (va > vb)) { s[i] = vb; s[p] = va; }
            }
            __syncthreads();
        }
    }
    for (unsigned e = threadIdx.x; e < TILE; e += blockDim.x)
        base[gbase + e] = s[e];
}

// ---------------------------------------------------------------------------
// K4b: one global bitonic compare-swap pass (only used while j >= TILE).
// ---------------------------------------------------------------------------
__global__ void k_bitonic(unsigned* __restrict__ keys, unsigned j, unsigned k)
{
    unsigned i   = blockIdx.x * blockDim.x + threadIdx.x;   // [0, PADN)
    unsigned c   = blockIdx.y;
    unsigned ixj = i ^ j;
    if (ixj > i) {
        unsigned* base = keys + (size_t)c * PADN;
        unsigned va = base[i];
        unsigned vb = base[ixj];
        bool desc = ((i & k) == 0);
        bool sw   = desc ? (va < vb) : (va > vb);
        if (sw) { base[i] = vb; base[ixj] = va; }
    }
}

// ---------------------------------------------------------------------------
// K4c: fused LDS merge: remaining passes j = TILE/2 .. 1 of stage k.
// ---------------------------------------------------------------------------
__global__ void k_bitonic_local_merge(unsigned* __restrict__ keys, unsigned k)
{
    const unsigned c     = blockIdx.y;
    const unsigned gbase = blockIdx.x * TILE;
    unsigned* base = keys + (size_t)c * PADN;
    __shared__ unsigned s[TILE];
    for (unsigned e = threadIdx.x; e < TILE; e += blockDim.x)
        s[e] = base[gbase + e];
    __syncthreads();
    for (unsigned j = TILE >> 1; j > 0; j >>= 1) {
        for (unsigned t = threadIdx.x; t < TILE / 2; t += blockDim.x) {
            unsigned i = ((t & ~(j - 1)) << 1) | (t & (j - 1));
            unsigned p = i | j;
            unsigned va = s[i], vb = s[p];
            bool desc = (((gbase + i) & k) == 0);
            if (desc ? (va < vb) : (va > vb)) { s[i] = vb; s[p] = va; }
        }
        __syncthreads();
    }
    for (unsigned e = threadIdx.x; e < TILE; e += blockDim.x)
        base[gbase + e] = s[e];
}

// ---------------------------------------------------------------------------
// K5: per-block gt counts over the sorted keys.
// ---------------------------------------------------------------------------
__global__ void k_gt_partials(const unsigned* __restrict__ keys,
                              unsigned*       __restrict__ gtp)
{
    const int c = blockIdx.y, b = blockIdx.x, tid = threadIdx.x;
    unsigned key = keys[(size_t)c * PADN + b * 256 + tid];
    __shared__ unsigned red[256];
    red[tid] = key & 1u;
    __syncthreads();
    for (int off = 128; off > 0; off >>= 1) {
        if (tid < off) red[tid] += red[tid + off];
        __syncthreads();
    }
    if (tid == 0) gtp[c * NBLK + b] = red[0];
}

// ---------------------------------------------------------------------------
// K6: sequential exclusive scan of per-block gt counts (exact integers).
// ---------------------------------------------------------------------------
__global__ void k_scan_partials(const unsigned* __restrict__ gtp,
                                unsigned*       __restrict__ basep)
{
    const int c = blockIdx.x;
    if (threadIdx.x == 0) {
        unsigned run = 0;
        for (int b = 0; b < NBLK; ++b) {
            basep[c * NBLK + b] = run;
            run += gtp[c * NBLK + b];
        }
    }
}

// ---------------------------------------------------------------------------
// K7: lovasz contribution per sorted element using exact inclusive cumsum of
// gt (block base + LDS Hillis-Steele scan). Padding keys have err=0 -> 0.
// ---------------------------------------------------------------------------
__global__ void k_lovasz_contrib(const unsigned* __restrict__ keys,
                                 const unsigned* __restrict__ basep,
                                 const float*    __restrict__ wsf,
                                 float*          __restrict__ lpart)
{
    const int c = blockIdx.y, b = blockIdx.x, tid = threadIdx.x;
    unsigned key = keys[(size_t)c * PADN + b * 256 + tid];
    unsigned gt  = key & 1u;
    float    err = __uint_as_float(key >> 1);

    __shared__ unsigned sg[256];
    sg[tid] = gt;
    __syncthreads();
    for (int off = 1; off < 256; off <<= 1) {
        unsigned v = (tid >= off) ? sg[tid - off] : 0u;
        __syncthreads();
        sg[tid] += v;
        __syncthreads();
    }

    float    P  = wsf[OFF_CNTF + c];
    unsigned Gi = basep[c * NBLK + b] + sg[tid];     // inclusive cumsum(gt)
    unsigned i  = (unsigned)b * 256u + (unsigned)tid;

    float Gf = (float)Gi;
    float Ji = 1.0f - (P - Gf) / (P + (float)(i + 1) - Gf);
    float Jm;
    if (i == 0) {
        Jm = 0.0f;
    } else {
        float Gp = (float)(Gi - gt);
        Jm = 1.0f - (P - Gp) / (P + (float)i - Gp);
    }
    float contrib = err * (Ji - Jm);                 // relu(err)=err, err>=0

    __shared__ float rf[256];
    rf[tid] = contrib;
    __syncthreads();
    for (int off = 128; off > 0; off >>= 1) {
        if (tid < off) rf[tid] += rf[tid + off];
        __syncthreads();
    }
    if (tid == 0) lpart[c * NBLK + b] = rf[0];
}

// ---------------------------------------------------------------------------
// K8: single-block deterministic finalize: seed loss, smoothing loss,
// lovasz partial reduction, total.
// ---------------------------------------------------------------------------
__global__ void k_finalize(const float* __restrict__ marg,
                           const float* __restrict__ seedn,
                           const int*   __restrict__ lab,
                           const float* __restrict__ wsf,
                           float*       __restrict__ out)
{
    const int tid = threadIdx.x;
    const float* probs = wsf + OFF_PROBS;
    const float* sigma = wsf + OFF_SIGMA;
    const float* cntf  = wsf + OFF_CNTF;
    const float* lpart = wsf + OFF_LPART;

    float seedAcc = 0.0f;
    float sm[NCLUST];
    for (int c = 0; c < NCLUST; ++c) sm[c] = 0.0f;
    for (int n = tid; n < NPTS; n += blockDim.x) {
        __builtin_prefetch(&probs[n + 8 * 256], 0, 1);
        float p  = probs[n];
        float pt = (p < 0.5f) ? 0.0f : p;            // thresholded detached prob
        seedAcc += fabsf(pt - seedn[n]);
        int l = lab[n];
        sm[l] += fabsf(marg[n] - sigma[l]);
    }

    __shared__ float rf[256];
    rf[tid] = seedAcc;
    __syncthreads();
    for (int off = 128; off > 0; off >>= 1) {
        if (tid < off) rf[tid] += rf[tid + off];
        __syncthreads();
    }
    float seedSum = rf[0];
    __syncthreads();

    float smoothAcc = 0.0f;                          // thread 0 only
    for (int c = 0; c < NCLUST; ++c) {
        rf[tid] = sm[c];
        __syncthreads();
        for (int off = 128; off > 0; off >>= 1) {
            if (tid < off) rf[tid] += rf[tid + off];
            __syncthreads();
        }
        if (tid == 0) smoothAcc += rf[0] / cntf[c];
        __syncthreads();
    }

    float maskAcc = 0.0f;                            // thread 0 only
    for (int c = 0; c < NCLUST; ++c) {
        float local = 0.0f;
        for (int b = tid; b < NBLK; b += blockDim.x) local += lpart[c * NBLK + b];
        rf[tid] = local;
        __syncthreads();
        for (int off = 128; off > 0; off >>= 1) {
            if (tid < off) rf[tid] += rf[tid + off];
            __syncthreads();
        }
        if (tid == 0) maskAcc += rf[0];
        __syncthreads();
    }

    if (tid == 0) {
        float reg        = wsf[OFF_REG];
        float mask_loss  = maskAcc / (float)NCLUST + 1e-4f * reg;
        float seed_loss  = seedSum / (float)NPTS;
        float smooth     = smoothAcc / (float)NCLUST;
        out[0] = 10.0f * mask_loss + 1.0f * seed_loss + 1.0f * smooth;
    }
}

// ---------------------------------------------------------------------------
extern "C" void kernel_launch(void* const* d_in, const int* in_sizes, int n_in,
                              void* d_out, int out_size, void* d_ws, size_t ws_size,
                              hipStream_t stream)
{
    (void)in_sizes; (void)n_in; (void)out_size; (void)ws_size;
    const float* emb   = (const float*)d_in[0];
    const float* marg  = (const float*)d_in[1];
    const float* seedn = (const float*)d_in[2];
    const int*   lab   = (const int*)d_in[3];
    float*    wsf = (float*)d_ws;
    unsigned* wsu = (unsigned*)d_ws;
    float*    out = (float*)d_out;

    // 1) per-cluster stats
    k_cluster_stats<<<NCLUST, 256, 0, stream>>>(emb, marg, lab, wsf);
    // 2) centroids / sigma / reg
    k_finalize_stats<<<1, 64, 0, stream>>>(wsf);
    // 3) WMMA distance tiles -> probs + packed sort keys (incl. zero padding)
    k_wmma_probs<<<PADN / 16 / 8, 256, 0, stream>>>(emb, lab, wsf, wsu);

    // 4) bitonic sort, descending, 16 clusters batched in grid.y.
    //    Stages k=2..TILE fused into one LDS kernel; for each larger stage,
    //    global passes only while j >= TILE, then one fused LDS merge for the
    //    inner passes. 55 launches total instead of 190 full L2 sweeps.
    unsigned* keys = wsu + OFF_KEYS;
    k_bitonic_local_sort<<<dim3(PADN / TILE, NCLUST), 256, 0, stream>>>(keys);
    for (unsigned k = TILE * 2; k <= PADN; k <<= 1) {
        for (unsigned j = k >> 1; j >= TILE; j >>= 1)
            k_bitonic<<<dim3(PADN / 256, NCLUST), 256, 0, stream>>>(keys, j, k);
        k_bitonic_local_merge<<<dim3(PADN / TILE, NCLUST), 256, 0, stream>>>(keys, k);
    }

    // 5) exact cumsum scaffolding + lovasz contributions
    unsigned* gtp   = wsu + OFF_GTP;
    unsigned* basep = wsu + OFF_BASE;
    float*    lpart = wsf + OFF_LPART;
    k_gt_partials<<<dim3(NBLK, NCLUST), 256, 0, stream>>>(keys, gtp);
    k_scan_partials<<<NCLUST, 32, 0, stream>>>(gtp, basep);
    k_lovasz_contrib<<<dim3(NBLK, NCLUST), 256, 0, stream>>>(keys, basep, wsf, lpart);
    // 6) final scalar
    k_finalize<<<1, 256, 0, stream>>>(marg, seedn, lab, wsf, out);
}